// Smm_88656714924904
// MI455X (gfx1250) — compile-verified
//
#include <hip/hip_runtime.h>
#include <hip/hip_bf16.h>
#include <climits>

#define TILE  256
#define BLOCK 256

// ---- gfx1250 async global->LDS path (probe via __has_builtin; safe fallback) ----
#if defined(__has_builtin)
#if __has_builtin(__builtin_amdgcn_global_load_async_to_lds_b32) && \
    __has_builtin(__builtin_amdgcn_s_wait_asynccnt)
#define HAVE_ASYNC 1
#endif
#endif
#ifndef HAVE_ASYNC
#define HAVE_ASYNC 0
#endif

typedef __attribute__((address_space(1))) int as1_int;
typedef __attribute__((address_space(3))) int as3_int;

__device__ __forceinline__ void async_copy_b32(const int* g, int* l) {
#if HAVE_ASYNC
  __builtin_amdgcn_global_load_async_to_lds_b32(
      (as1_int*)const_cast<int*>(g),
      (as3_int*)l,
      /*offset=*/0, /*cpol=*/0);
#else
  *l = *g;   // fallback: sync load + DS store
#endif
}

template <int N>
__device__ __forceinline__ void wait_async() {
#if HAVE_ASYNC
  __builtin_amdgcn_s_wait_asynccnt(N);
#endif
}

// ---- kernels -------------------------------------------------------------

__global__ void k_init_slot(int* __restrict__ slot, int N) {
  int i = blockIdx.x * blockDim.x + threadIdx.x;
  if (i < N) slot[i] = INT_MAX;
}

__global__ void k_mark(const int* __restrict__ perm, int* __restrict__ slot, int K) {
  int k = blockIdx.x * blockDim.x + threadIdx.x;
  if (k < K) atomicMin(&slot[perm[k]], k);
}

// out[k][:] = x[perm[k]][:]  (self-loop term), float4 vectorized
__global__ void k_init_out(const float4* __restrict__ x4, const int* __restrict__ perm,
                           float4* __restrict__ out4, int K, int C4) {
  int i = blockIdx.x * blockDim.x + threadIdx.x;
  if (i < K * C4) {
    int k = i / C4;
    int j = i - k * C4;
    out4[i] = x4[(size_t)perm[k] * C4 + j];
  }
}

// Edge scatter-add: wave32 per edge (32 lanes x float4 = 128 channels),
// edge indices double-buffer-staged into LDS via async loads.
__global__ void __launch_bounds__(BLOCK) k_edges(
    const float* __restrict__ x, const int* __restrict__ rows,
    const int* __restrict__ cols, const int* __restrict__ slot,
    float* __restrict__ out, int E, int C) {
  __shared__ int s_row[2][TILE];
  __shared__ int s_col[2][TILE];

  const int tid  = threadIdx.x;
  const int lane = tid & 31;
  const int wave = tid >> 5;       // 8 waves per block
  const int C4   = C >> 2;
  const float4* x4 = (const float4*)x;
  const int nt = (E + TILE - 1) / TILE;

  auto load_tile = [&](int t, int b) {
    int idx = t * TILE + tid;
    if (idx > E - 1) idx = E - 1;  // clamp tail (garbage lanes never processed)
    async_copy_b32(&rows[idx], &s_row[b][tid]);
    async_copy_b32(&cols[idx], &s_col[b][tid]);
  };

  int tile = blockIdx.x;
  int buf = 0;
  if (tile < nt) load_tile(tile, 0);

  for (; tile < nt; tile += gridDim.x) {
    int next = tile + gridDim.x;
    if (next < nt) { load_tile(next, buf ^ 1); wait_async<2>(); }  // prev tile landed
    else           { wait_async<0>(); }
    __syncthreads();

    const int base = tile * TILE;

    // Warm L2/L0 for the 32 gather rows this wave is about to walk.
    {
      int e = (wave << 5) | lane;
      if (base + e < E) {
        int c = s_col[buf][e];
        __builtin_prefetch(&x[(size_t)c * C], 0, 1);   // -> global_prefetch_b8
      }
    }

    for (int j = 0; j < 32; ++j) {
      int e  = (wave << 5) + j;
      int ge = base + e;
      if (ge >= E) break;
      int r = s_row[buf][e];       // LDS broadcast read
      int c = s_col[buf][e];
      int s = slot[r];             // 200 KB table, L2-resident
      if (s != INT_MAX && r != c) {
        for (int ch = lane; ch < C4; ch += 32) {
          float4 v = x4[(size_t)c * C4 + ch];          // coalesced 512B row gather
          float* dst = out + (size_t)s * C + (ch << 2);
          atomicAdd(dst + 0, v.x);                     // global_atomic_add_f32 (L2)
          atomicAdd(dst + 1, v.y);
          atomicAdd(dst + 2, v.z);
          atomicAdd(dst + 3, v.w);
        }
      }
    }
    __syncthreads();
    buf ^= 1;
  }
}

// Duplicate perm entries: copy canonical accumulated row.
__global__ void k_dups(const int* __restrict__ perm, const int* __restrict__ slot,
                       float4* __restrict__ out4, int K, int C4) {
  int wid  = (int)((blockIdx.x * blockDim.x + threadIdx.x) >> 5);
  int lane = threadIdx.x & 31;
  if (wid >= K) return;
  int s = slot[perm[wid]];
  if (s == wid) return;            // canonical row: already complete
  for (int ch = lane; ch < C4; ch += 32)
    out4[(size_t)wid * C4 + ch] = out4[(size_t)s * C4 + ch];
}

// ---- entry ---------------------------------------------------------------

extern "C" void kernel_launch(void* const* d_in, const int* in_sizes, int n_in,
                              void* d_out, int out_size, void* d_ws, size_t ws_size,
                              hipStream_t stream) {
  const float* x     = (const float*)d_in[0];
  const int*   edges = (const int*)d_in[1];   // (2, E): rows then cols
  const int*   perm  = (const int*)d_in[2];
  float*       out   = (float*)d_out;

  const int K  = in_sizes[2];
  const int E  = in_sizes[1] / 2;
  const int C  = out_size / K;        // 128
  const int N  = in_sizes[0] / C;     // 50000
  const int C4 = C / 4;

  int* slot = (int*)d_ws;             // N ints = 200 KB

  const int* rows = edges;
  const int* cols = edges + E;

  k_init_slot<<<(N + 255) / 256, 256, 0, stream>>>(slot, N);
  k_mark<<<(K + 255) / 256, 256, 0, stream>>>(perm, slot, K);
  k_init_out<<<(K * C4 + 255) / 256, 256, 0, stream>>>(
      (const float4*)x, perm, (float4*)out, K, C4);

  int nt = (E + TILE - 1) / TILE;
  int blocks = nt < 1024 ? nt : 1024;
  k_edges<<<blocks, BLOCK, 0, stream>>>(x, rows, cols, slot, out, E, C);

  k_dups<<<(K * 32 + 255) / 256, 256, 0, stream>>>(perm, slot, (float4*)out, K, C4);
}